// Net_23828478558452
// MI455X (gfx1250) — compile-verified
//
#include <hip/hip_runtime.h>
#include <hip/hip_bf16.h>

// ---------------------------------------------------------------------------
// GCN link prediction on MI455X (gfx1250, wave32).
// GEMMs use V_WMMA_F32_16X16X4_F32 (exact fp32) since FLOPs are negligible
// vs the ~1.5GB of edge-scatter traffic; everything else is streaming +
// global f32 atomics.
// ---------------------------------------------------------------------------

typedef float v2f __attribute__((ext_vector_type(2)));
typedef float v8f __attribute__((ext_vector_type(8)));

#define F_IN  256
#define F_MID 64
#define F_OUT 32

// ------------------------------ init helpers ------------------------------

__global__ void __launch_bounds__(256) fill_kernel(float* __restrict__ p, float v, int n) {
    int i = blockIdx.x * 256 + threadIdx.x;
    if (i < n) p[i] = v;
}

__global__ void __launch_bounds__(256) deg_count_kernel(const int* __restrict__ dst,
                                                        float* __restrict__ deg, int n_edges) {
    int e = blockIdx.x * 256 + threadIdx.x;
    if (e < n_edges) atomicAdd(&deg[dst[e]], 1.0f);
}

__global__ void __launch_bounds__(256) rsqrt_kernel(float* __restrict__ d, int n) {
    int i = blockIdx.x * 256 + threadIdx.x;
    if (i < n) d[i] = rsqrtf(d[i]);   // deg >= 1 always (self loop)
}

// ------------------------------ WMMA GEMM -------------------------------
// h[N,NOUT] = x[N,K] @ W[K,NOUT].  8 waves/block, each wave: 16 rows x NOUT cols.
// W is staged into LDS *transposed* with a padded stride (K+4) so the B
// fragment is a single contiguous ds_load_b64 and the two lane-halves of the
// wave (K offsets kk and kk+2) land on disjoint LDS banks: bank(lower)=4l+kk,
// bank(upper)=4l+kk+2 -> all 64 banks used exactly once, zero conflicts.

template <int K, int NOUT>
__global__ void __launch_bounds__(256) gemm_kernel(const float* __restrict__ x,
                                                   const float* __restrict__ W,
                                                   float* __restrict__ h, int n_nodes) {
    constexpr int LDK = K + 4;                 // padded stride (words)
    constexpr int NT  = NOUT / 16;             // N-tiles per wave
    __shared__ float sWt[NOUT * LDK];          // transposed: sWt[n][k]

    for (int i = threadIdx.x; i < K * NOUT; i += 256) {
        const int k = i / NOUT;
        const int n = i % NOUT;
        sWt[n * LDK + k] = W[i];
    }
    __syncthreads();

    const int wave = threadIdx.x >> 5;
    const int lane = threadIdx.x & 31;
    const int half = lane >> 4;        // 0: K=k,k+1   1: K=k+2,k+3
    const int l    = lane & 15;
    const int m0   = (blockIdx.x * 8 + wave) * 16;
    if (m0 >= n_nodes) return;

    const int rowA = min(m0 + l, n_nodes - 1);         // clamp loads at tail
    const float* __restrict__ xrow = x + (size_t)rowA * K;

    v8f acc[NT] = {};
    for (int k = 0; k < K; k += 4) {
        const int kk = k + 2 * half;
        const v2f a = *(const v2f*)(xrow + kk);        // 8B-aligned (kk even)
#pragma unroll
        for (int t = 0; t < NT; ++t) {
            const int col = t * 16 + l;
            const v2f b = *(const v2f*)(&sWt[col * LDK + kk]);
            acc[t] = __builtin_amdgcn_wmma_f32_16x16x4_f32(
                false, a, false, b, (short)0, acc[t], false, false);
        }
    }

    // C layout: lane half h, sub-lane l -> row m0 + 8*h + v, col t*16 + l.
    if (m0 + 16 <= n_nodes) {                          // fast path: no guards
        float* __restrict__ hp = h + (size_t)(m0 + 8 * half) * NOUT + l;
#pragma unroll
        for (int v = 0; v < 8; ++v) {
#pragma unroll
            for (int t = 0; t < NT; ++t)
                hp[(size_t)v * NOUT + t * 16] = acc[t][v];
        }
    } else {                                           // ragged tail block
#pragma unroll
        for (int v = 0; v < 8; ++v) {
            const int row = m0 + 8 * half + v;
            if (row < n_nodes) {
#pragma unroll
                for (int t = 0; t < NT; ++t)
                    h[(size_t)row * NOUT + t * 16 + l] = acc[t][v];
            }
        }
    }
}

// ------------------------------ edge scatter ------------------------------
// agg[dst] += dinv[src]*dinv[dst] * h[src]   (one thread per (edge, feature))

template <int F>
__global__ void __launch_bounds__(256) scatter_kernel(const int* __restrict__ src,
                                                      const int* __restrict__ dst,
                                                      const float* __restrict__ dinv,
                                                      const float* __restrict__ h,
                                                      float* __restrict__ agg, int n_edges) {
    const int idx = blockIdx.x * 256 + threadIdx.x;
    const int e = idx / F;
    const int f = idx % F;
    if (e >= n_edges) return;
    const int s = src[e];
    const int d = dst[e];
    const float norm = dinv[s] * dinv[d];
    atomicAdd(&agg[(size_t)d * F + f], norm * h[(size_t)s * F + f]);
}

// self-loop contribution + bias (+ optional ReLU), in place on agg.
template <int F, bool RELU>
__global__ void __launch_bounds__(256) finalize_kernel(float* __restrict__ agg,
                                                       const float* __restrict__ h,
                                                       const float* __restrict__ dinv,
                                                       const float* __restrict__ bias,
                                                       int n_nodes) {
    const int idx = blockIdx.x * 256 + threadIdx.x;
    if (idx >= n_nodes * F) return;
    const int i = idx / F;
    const int f = idx % F;
    const float di = dinv[i];
    float v = agg[idx] + di * di * h[idx] + bias[f];
    if (RELU) v = fmaxf(v, 0.0f);
    agg[idx] = v;
}

// ------------------------------ link decode -------------------------------
// logits[k] = dot32(z[a], z[b]) for pos edges then neg edges. 1 wave / edge.

__global__ void __launch_bounds__(256) decode_kernel(const int* __restrict__ pos,
                                                     const int* __restrict__ neg,
                                                     const float* __restrict__ z,
                                                     float* __restrict__ out, int n_eval) {
    const int gw   = (blockIdx.x * 256 + threadIdx.x) >> 5;  // global wave id
    const int lane = threadIdx.x & 31;
    if (gw >= 2 * n_eval) return;
    int a, b;
    if (gw < n_eval) { a = pos[gw];            b = pos[n_eval + gw]; }
    else             { int k = gw - n_eval; a = neg[k]; b = neg[n_eval + k]; }
    float v = z[(size_t)a * F_OUT + lane] * z[(size_t)b * F_OUT + lane];
#pragma unroll
    for (int o = 16; o > 0; o >>= 1) v += __shfl_xor(v, o, 32);
    if (lane == 0) out[gw] = v;
}

// ------------------------------ launcher ----------------------------------

static inline size_t align_up(size_t x) { return (x + 255) & ~(size_t)255; }

extern "C" void kernel_launch(void* const* d_in, const int* in_sizes, int n_in,
                              void* d_out, int out_size, void* d_ws, size_t ws_size,
                              hipStream_t stream) {
    const float* x   = (const float*)d_in[0];
    const float* W1  = (const float*)d_in[1];
    const float* b1  = (const float*)d_in[2];
    const float* W2  = (const float*)d_in[3];
    const float* b2  = (const float*)d_in[4];
    const int*   ei  = (const int*)d_in[5];
    const int*   pos = (const int*)d_in[6];
    const int*   neg = (const int*)d_in[7];
    float* out = (float*)d_out;

    const int n_nodes = in_sizes[0] / F_IN;
    const int n_edges = in_sizes[5] / 2;
    const int n_eval  = in_sizes[6] / 2;
    const int* src = ei;
    const int* dst = ei + n_edges;

    // workspace partition
    char* w = (char*)d_ws;
    size_t off = 0;
    float* dinv = (float*)(w + off); off += align_up((size_t)n_nodes * 4);
    float* h1   = (float*)(w + off); off += align_up((size_t)n_nodes * F_MID * 4);
    float* agg1 = (float*)(w + off); off += align_up((size_t)n_nodes * F_MID * 4);
    float* h2   = (float*)(w + off); off += align_up((size_t)n_nodes * F_OUT * 4);
    float* z    = (float*)(w + off); off += align_up((size_t)n_nodes * F_OUT * 4);
    (void)ws_size;

    const int B = 256;
    const int nm = n_nodes * F_MID;   // 6.4M
    const int no = n_nodes * F_OUT;   // 3.2M

    // 1) degree (self-loop included) -> dinv
    fill_kernel<<<(n_nodes + B - 1) / B, B, 0, stream>>>(dinv, 1.0f, n_nodes);
    fill_kernel<<<(nm + B - 1) / B, B, 0, stream>>>(agg1, 0.0f, nm);
    fill_kernel<<<(no + B - 1) / B, B, 0, stream>>>(z, 0.0f, no);
    deg_count_kernel<<<(n_edges + B - 1) / B, B, 0, stream>>>(dst, dinv, n_edges);
    rsqrt_kernel<<<(n_nodes + B - 1) / B, B, 0, stream>>>(dinv, n_nodes);

    // 2) layer 1: GEMM -> scatter -> self-loop+bias+relu
    const int gemm_blocks = (n_nodes + 128 - 1) / 128;     // 8 waves x 16 rows
    gemm_kernel<F_IN, F_MID><<<gemm_blocks, B, 0, stream>>>(x, W1, h1, n_nodes);
    {
        long long t = (long long)n_edges * F_MID;
        scatter_kernel<F_MID><<<(unsigned)((t + B - 1) / B), B, 0, stream>>>(
            src, dst, dinv, h1, agg1, n_edges);
    }
    finalize_kernel<F_MID, true><<<(nm + B - 1) / B, B, 0, stream>>>(agg1, h1, dinv, b1, n_nodes);

    // 3) layer 2: GEMM -> scatter -> self-loop+bias
    gemm_kernel<F_MID, F_OUT><<<gemm_blocks, B, 0, stream>>>(agg1, W2, h2, n_nodes);
    {
        long long t = (long long)n_edges * F_OUT;
        scatter_kernel<F_OUT><<<(unsigned)((t + B - 1) / B), B, 0, stream>>>(
            src, dst, dinv, h2, z, n_edges);
    }
    finalize_kernel<F_OUT, false><<<(no + B - 1) / B, B, 0, stream>>>(z, h2, dinv, b2, n_nodes);

    // 4) decode: one wave32 per query edge
    {
        int waves = 2 * n_eval;
        int blocks = (waves * 32 + B - 1) / B;
        decode_kernel<<<blocks, B, 0, stream>>>(pos, neg, z, out, n_eval);
    }
    (void)out_size; (void)n_in;
}